// CBOW_HS_55130200212125
// MI455X (gfx1250) — compile-verified
//
#include <hip/hip_runtime.h>

// CBOW hierarchical-softmax loss, MI455X (gfx1250, wave32).
// Gather-bound; both embedding tables are L2-resident (51.2 MB each < 192 MB L2).
// Batched dot products done on the matrix pipe via V_WMMA_F32_16X16X4_F32
// (full f32 precision), extracting the diagonal of V·U^T for 16 examples/wave.

typedef __attribute__((ext_vector_type(2))) float v2f;
typedef __attribute__((ext_vector_type(4))) float v4f;
typedef __attribute__((ext_vector_type(8))) float v8f;

constexpr int Bsz  = 65536;  // batch
constexpr int Cn   = 10;     // context words
constexpr int Dn   = 18;     // tree depth
constexpr int En   = 128;    // embedding dim
constexpr int WAVES = 4;     // waves per block
constexpr int TILE  = 16;    // examples per wave (== WMMA M/N)
constexpr int VROW  = 132;   // padded LDS row stride (floats): bank-conflict-free
constexpr float EPSf = 1e-9f;

__global__ __launch_bounds__(WAVES * 32)
void cbow_hs_wmma_kernel(const int* __restrict__ ctx,         // [B, C]
                         const int* __restrict__ path,        // [B, D]
                         const int* __restrict__ codes,       // [B, D]
                         const float* __restrict__ in_embed,  // [V, E]
                         const float* __restrict__ node_embed,// [N, E]
                         float* __restrict__ out)             // [B]
{
    __shared__ float vsh[WAVES * TILE * VROW];   // 33,792 B

    const int lane = threadIdx.x & 31;
    const int wave = threadIdx.x >> 5;
    const int tileBase = (blockIdx.x * WAVES + wave) * TILE;
    float* vtile = vsh + wave * TILE * VROW;

    // ---------- Phase 1: context means -> LDS ----------
    // One wave covers E=128 with a float4 per lane; fully coalesced 16B loads.
    for (int b = 0; b < TILE; ++b) {
        const int ex = tileBase + b;
        v4f acc = {0.f, 0.f, 0.f, 0.f};
        #pragma unroll
        for (int c = 0; c < Cn; ++c) {
            const int idx = ctx[ex * Cn + c];
            acc += *(const v4f*)(in_embed + (size_t)idx * En + lane * 4);
        }
        acc *= (1.0f / (float)Cn);
        *(v4f*)(vtile + b * VROW + lane * 4) = acc;
    }
    __syncthreads();

    // ---------- Phase 2: logits via f32 WMMA diagonal, fused loss ----------
    // A(16x4 f32) layout: lanes 0-15 -> M=lane, {V0,V1}={K0,K1}; lanes 16-31 -> {K2,K3}
    // B(4x16 f32) layout: N=lane&15;  lanes 0-15 -> {K0,K1};      lanes 16-31 -> {K2,K3}
    const int laneM = lane & 15;
    const int kHalf = (lane >> 4) << 1;                 // 0 or 2
    const float* vrow = vtile + laneM * VROW + kHalf;   // A fragment source (LDS)
    const int exN = tileBase + laneM;                   // example this lane feeds/owns

    float lossAcc = 0.f;
    const bool active = (lane < 8) || (lane >= 24);     // lanes holding diagonal elems

    int node = path[exN * Dn];                          // software-pipelined node index
    #pragma unroll 1
    for (int d = 0; d < Dn; ++d) {
        const int nodeNext = (d + 1 < Dn) ? path[exN * Dn + d + 1] : node;
        const float* urow = node_embed + (size_t)node * En + kHalf;
        __builtin_prefetch(node_embed + (size_t)nodeNext * En, 0, 1);

        v8f acc = {};
        #pragma unroll
        for (int kc = 0; kc < En / 4; ++kc) {
            v2f a = *(const v2f*)(vrow + 4 * kc);        // ds_load_b64, conflict-free
            v2f b = *(const v2f*)(urow + 4 * kc);        // global_load_b64 gather (L2)
            // D[m,n] += sum_k V[m,k] * U[n,k]; diag D[b,b] = logit of example b
            acc = __builtin_amdgcn_wmma_f32_16x16x4_f32(
                /*neg_a=*/false, a, /*neg_b=*/false, b,
                /*c_mod=*/(short)0, acc, /*reuse_a=*/false, /*reuse_b=*/false);
        }

        // Branch-free diagonal extraction (EXEC stays all-ones around WMMAs):
        // D[b,b]: b<8 -> lane b, vgpr b ; b>=8 -> lane 16+b, vgpr b-8.
        float logit = 0.f;
        #pragma unroll
        for (int g = 0; g < 8; ++g) {
            const bool sel = (lane == g) || (lane == 24 + g);
            logit = sel ? acc[g] : logit;
        }

        const int code = codes[exN * Dn + d];            // harmless on inactive lanes
        const float s  = 1.f / (1.f + __expf(-logit));
        const float p  = (code == 1) ? s : (1.f - s);
        const float term = -__logf(p + EPSf);
        lossAcc += active ? term : 0.f;

        node = nodeNext;
    }

    // lane b (b<8) and lane 16+b (b>=8) own example laneM's loss
    if (active) out[tileBase + laneM] = lossAcc;
}

extern "C" void kernel_launch(void* const* d_in, const int* in_sizes, int n_in,
                              void* d_out, int out_size, void* d_ws, size_t ws_size,
                              hipStream_t stream) {
    const int*   ctx        = (const int*)d_in[0];
    const int*   path       = (const int*)d_in[1];
    const int*   codes      = (const int*)d_in[2];
    const float* in_embed   = (const float*)d_in[3];
    const float* node_embed = (const float*)d_in[4];
    float*       out        = (float*)d_out;

    dim3 grid(Bsz / (WAVES * TILE));   // 65536 / 64 = 1024
    dim3 block(WAVES * 32);            // 128 threads = 4 waves
    hipLaunchKernelGGL(cbow_hs_wmma_kernel, grid, block, 0, stream,
                       ctx, path, codes, in_embed, node_embed, out);
}